// ItemCode_12575664243197
// MI455X (gfx1250) — compile-verified
//
#include <hip/hip_runtime.h>
#include <hip/hip_bf16.h>
#include <stdint.h>

// Problem constants (from reference)
#define PQ_M_   8
#define SUB_    64          // floats per sub-vector
#define EMB_    512         // floats per token embedding
#define TOKENS_ (256 * 512) // BATCH * SEQ

// Cache policy immediates (gfx12+ CPol encoding: TH in bits [2:0], scope [4:3]).
// Output is a 268 MB write-once stream (> 192 MB L2) -> non-temporal stores so
// the 512 KB centroid table stays L2-resident for the gather loads.
#define CPOL_RT      0
#define CPOL_ST_NT   1

// CDNA5 async global<->LDS path (ASYNCcnt-tracked), guarded so the file
// compiles on host pass / older toolchains with a plain-VGPR fallback.
#if __has_builtin(__builtin_amdgcn_global_load_async_to_lds_b128) && \
    __has_builtin(__builtin_amdgcn_global_store_async_from_lds_b128)
#define USE_ASYNC 1
// Probe-learned signature: b128 async builtins take (AS1 v4i*, AS3 v4i*, imm, imm cpol).
typedef int v4i __attribute__((__vector_size__(4 * sizeof(int))));
typedef __attribute__((address_space(1))) v4i GV4; // global int4
typedef __attribute__((address_space(3))) v4i LV4; // LDS int4
#else
#define USE_ASYNC 0
#endif

__global__ __launch_bounds__(256) void pq_gather_kernel(
    const int* __restrict__ input_ids,   // [TOKENS_]
    const int* __restrict__ item_codes,  // [NUM_ITEMS, 8]
    const float* __restrict__ centroids, // [8, 256, 64]
    float* __restrict__ out)             // [TOKENS_, 512]
{
    const int lane = threadIdx.x & 31;
    const int wave = threadIdx.x >> 5;
    const int t    = blockIdx.x * 8 + wave;   // one wave per token

    const int id = input_ids[t];              // wave-uniform
    float* dst = out + (size_t)t * EMB_;

    if (id == 0) {
        // padding token: zero the whole 512-float embedding, coalesced b128 stores
        const float4 z = make_float4(0.f, 0.f, 0.f, 0.f);
#pragma unroll
        for (int j = 0; j < 4; ++j)
            reinterpret_cast<float4*>(dst)[j * 32 + lane] = z;
        return;
    }

    // Each lane loads one of the 8 codes; broadcast later via wave32 shuffle.
    int c = item_codes[(size_t)id * PQ_M_ + (lane & 7)];
    c = c < 255 ? c : 255; // jnp.minimum(codes, 255)

#if USE_ASYNC
    // 2 KB staging slice per wave; data flows memory -> LDS -> memory without VGPRs.
    __shared__ __align__(16) char lds_buf[8 * 2048];
    char* my = lds_buf + wave * 2048;

#pragma unroll
    for (int j = 0; j < 4; ++j) {
        const int e4 = j * 32 + lane;        // float4 index within the 512-float emb
        const int m  = e4 >> 4;              // sub-dim (64 floats = 16 float4 per m)
        const int cm = __shfl(c, m, 32);     // code for this sub-dim
        const float* src = centroids
            + ((size_t)(m * 256 + cm)) * SUB_   // row base
            + (size_t)(e4 & 15) * 4;            // offset within 64-float row
        __builtin_amdgcn_global_load_async_to_lds_b128(
            (GV4*)src, (LV4*)(my + e4 * 16), 0, CPOL_RT);
    }

    // Ensure the async gathers have landed in LDS before the async stores read it.
#if __has_builtin(__builtin_amdgcn_s_wait_asynccnt)
    __builtin_amdgcn_s_wait_asynccnt(0);
#else
    asm volatile("s_wait_asynccnt 0" ::: "memory");
#endif

#pragma unroll
    for (int j = 0; j < 4; ++j) {
        const int e4 = j * 32 + lane;
        __builtin_amdgcn_global_store_async_from_lds_b128(
            (GV4*)(dst + (size_t)e4 * 4), (LV4*)(my + e4 * 16), 0, CPOL_ST_NT);
    }

    // Drain stores before wave end (S_ENDPGM implies wait-idle, but be explicit).
#if __has_builtin(__builtin_amdgcn_s_wait_asynccnt)
    __builtin_amdgcn_s_wait_asynccnt(0);
#else
    asm volatile("s_wait_asynccnt 0" ::: "memory");
#endif

#else // fallback: plain b128 load/store through VGPRs
#pragma unroll
    for (int j = 0; j < 4; ++j) {
        const int e4 = j * 32 + lane;
        const int m  = e4 >> 4;
        const int cm = __shfl(c, m, 32);
        const float4 v = reinterpret_cast<const float4*>(centroids)
            [(size_t)(m * 256 + cm) * 16 + (e4 & 15)];
        reinterpret_cast<float4*>(dst)[e4] = v;
    }
#endif
}

extern "C" void kernel_launch(void* const* d_in, const int* in_sizes, int n_in,
                              void* d_out, int out_size, void* d_ws, size_t ws_size,
                              hipStream_t stream) {
    (void)in_sizes; (void)n_in; (void)out_size; (void)d_ws; (void)ws_size;
    const int*   input_ids  = (const int*)d_in[0];
    const int*   item_codes = (const int*)d_in[1];
    const float* centroids  = (const float*)d_in[2];
    float*       out        = (float*)d_out;

    dim3 block(256);                 // 8 wave32 waves
    dim3 grid(TOKENS_ / 8);          // one wave per token, exact cover
    pq_gather_kernel<<<grid, block, 0, stream>>>(input_ids, item_codes, centroids, out);
}